// DeconvSlimCapsule3D_90598040142294
// MI455X (gfx1250) — compile-verified
//
#include <hip/hip_runtime.h>
#include <hip/hip_bf16.h>

typedef float v2f __attribute__((ext_vector_type(2)));
typedef float v8f __attribute__((ext_vector_type(8)));

// Problem constants (from reference): B=4, I=4, O=4, AI=16, AO=16, K=4, S=2, P=1
// Deconv out spatial = 32^3. votes[b,i,o,ao,od,oh,ow]; routing local per voxel.
//
// LDS layout (floats):
//   xs : [i(4)][tdi(2)][thi(2)][iw+1 (18, zero-padded ends)][ai(16)]  = 4608
//   ws : [tdi(2)][thi(2)][tw(4)][c=o*16+ao (64)][ai(16)]             = 16384
//   vsm: [par(2)][vox(16)][p=i*4+o (16)][ao(16)]                     = 8192
#define XS_N 4608
#define WS_N 16384
#define VS_N 8192

__global__ __launch_bounds__(1024)
void deconv_caps_routing(const float* __restrict__ xg,
                         const float* __restrict__ wg,
                         const float* __restrict__ bg,
                         float* __restrict__ outg) {
  extern __shared__ float smem[];
  float* xs  = smem;
  float* ws  = smem + XS_N;
  float* vsm = smem + XS_N + WS_N;

  const int tid = threadIdx.x;
  const int blk = blockIdx.x;
  const int b  = blk >> 10;        // batch
  const int od = (blk >> 5) & 31;  // out depth
  const int oh = blk & 31;         // out height

  const int tdbase = 1 - (od & 1); // allowed taps: tdbase + 2*tdi
  const int thbase = 1 - (oh & 1);
  const int id0 = (od + 1) >> 1;   // id = id0 - tdi
  const int ih0 = (oh + 1) >> 1;   // ih = ih0 - thi

  // ---- Stage weights slice into LDS: 16384 elems / 1024 threads ----
  for (int idx = tid; idx < WS_N; idx += 1024) {
    int ai  = idx & 15;
    int c   = (idx >> 4) & 63;
    int tw  = (idx >> 10) & 3;
    int thi = (idx >> 12) & 1;
    int tdi = (idx >> 13) & 1;
    int td = tdbase + 2 * tdi;
    int th = thbase + 2 * thi;
    // w[ai, c, td, th, tw] ; strides: ai=4096, c=64
    ws[idx] = wg[((ai * 64 + c) << 6) + td * 16 + th * 4 + tw];
  }
  // ---- Stage x slice into LDS (iw padded by 1 on each end with zeros) ----
  for (int idx = tid; idx < XS_N; idx += 1024) {
    int ai   = idx & 15;
    int t    = idx >> 4;      // 0..287
    int iwp  = t % 18;
    int rest = t / 18;        // 0..15 = ((i*2+tdi)*2+thi)
    int thi = rest & 1;
    int tdi = (rest >> 1) & 1;
    int ci  = rest >> 2;      // input capsule i
    int iw = iwp - 1;
    int id = id0 - tdi;
    int ih = ih0 - thi;
    float v = 0.f;
    if ((unsigned)iw < 16u && (unsigned)id < 16u && (unsigned)ih < 16u)
      v = xg[(((b * 4 + ci) * 16 + ai) << 12) + (id << 8) + (ih << 4) + iw];
    xs[idx] = v;
  }
  __syncthreads();

  // ---- GEMM: wave = (par, i, o); D[16 vox x 16 ao] += A[16x4] * B[4x16], K=128 ----
  const int lane = tid & 31;
  const int wv   = tid >> 5;       // 0..31
  const int par  = wv >> 4;        // ow parity
  const int icap = (wv >> 2) & 3;  // input capsule i
  const int ocap = wv & 3;         // output capsule o
  const int ln = lane & 15;        // A: row (voxel) ; B: col (ao) ; D: col (ao)
  const int kh = lane >> 4;        // K-half selector per ISA 16x4 f32 layout

  v8f acc = {0.f, 0.f, 0.f, 0.f, 0.f, 0.f, 0.f, 0.f};
#pragma unroll
  for (int j = 0; j < 32; ++j) {   // 32 chunks of K=4  (K = tap(8) x ai(16))
    int tap = j >> 2;              // uniform per chunk
    int twi = tap & 1;
    int thi = (tap >> 1) & 1;
    int tdi = tap >> 2;
    int aibase = ((j & 3) << 2) + (kh << 1);       // ai pair start
    int iw = ln + par - twi;                       // in [-1,16] -> padded idx iw+1
    int aoff = ((((icap * 2 + tdi) * 2 + thi) * 18 + (iw + 1)) << 4) + aibase;
    int tw = (1 - par) + (twi << 1);
    int boff = (((((tdi * 2 + thi) << 2) + tw) * 64 + (ocap * 16 + ln)) << 4) + aibase;
    v2f a  = *(const v2f*)(xs + aoff);
    v2f bb = *(const v2f*)(ws + boff);
    acc = __builtin_amdgcn_wmma_f32_16x16x4_f32(
        /*neg_a=*/false, a, /*neg_b=*/false, bb,
        /*c_mod=*/(short)0, acc, /*reuse_a=*/false, /*reuse_b=*/false);
  }

  // ---- Store votes tile to LDS.  D layout: vgpr g -> vox = g + 8*kh, col ao = ln ----
  const int p_idx = icap * 4 + ocap;
#pragma unroll
  for (int g = 0; g < 8; ++g) {
    int vx = g + (kh << 3);
    vsm[(((par * 16 + vx) * 16 + p_idx) << 4) + ln] = acc[g];
  }
  __syncthreads();

  // ---- Routing: one wave per voxel (rpar, rr). Lane = (p=(i,o) in bits0-3, h=bit4) ----
  const int rr = wv & 15;          // voxel index within parity
  const int p  = lane & 15;
  const int h  = lane >> 4;        // ao half: lanes handle ao in [8h, 8h+8)
  const int ri = p >> 2;           // i
  const int ro = p & 3;            // o

  float vv[8], bv[8];
  const float* vbase = vsm + (((par * 16 + rr) * 16 + p) << 4) + (h << 3);
#pragma unroll
  for (int j = 0; j < 8; ++j) vv[j] = vbase[j];
#pragma unroll
  for (int j = 0; j < 8; ++j) bv[j] = bg[ro * 16 + (h << 3) + j];

  float logit = 0.f;
  float pre[8];
  float scale = 0.f;
#pragma unroll
  for (int it = 0; it < 3; ++it) {
    // softmax over o (lane bits 0-1), replicated over h
    float m = logit;
    m = fmaxf(m, __shfl_xor(m, 1));
    m = fmaxf(m, __shfl_xor(m, 2));
    float e = __expf(logit - m);
    float s = e;
    s += __shfl_xor(s, 1);
    s += __shfl_xor(s, 2);
    float route = e / s;
    // pre[o][ao] = sum_i route*votes + bias   (sum over i = lane bits 2-3)
#pragma unroll
    for (int j = 0; j < 8; ++j) pre[j] = route * vv[j];
#pragma unroll
    for (int j = 0; j < 8; ++j) {
      pre[j] += __shfl_xor(pre[j], 4);
      pre[j] += __shfl_xor(pre[j], 8);
      pre[j] += bv[j];
    }
    if (it < 2) {
      float dt = 0.f, np2 = 0.f, nv2 = 0.f;
#pragma unroll
      for (int j = 0; j < 8; ++j) {
        dt  += pre[j] * vv[j];
        np2 += pre[j] * pre[j];
        nv2 += vv[j] * vv[j];
      }
      dt  += __shfl_xor(dt, 16);   // complete ao reduction across h
      np2 += __shfl_xor(np2, 16);
      nv2 += __shfl_xor(nv2, 16);
      float na = sqrtf(np2), nb = sqrtf(nv2);
      logit += dt / fmaxf(na * nb, 1e-8f);
    } else {
      // squash over ao
      float n2 = 0.f;
#pragma unroll
      for (int j = 0; j < 8; ++j) n2 += pre[j] * pre[j];
      n2 += __shfl_xor(n2, 16);
      float norm = sqrtf(n2);
      scale = n2 / ((norm + 1e-12f) * (1.f + n2));
    }
  }

  // ---- Write act[b, o, ao, od, oh, ow]; pre/scale replicated over i -> i==0 writes ----
  if (ri == 0) {
    int ow = (rr << 1) + par;
#pragma unroll
    for (int j = 0; j < 8; ++j) {
      int ao = (h << 3) + j;
      outg[(((b * 4 + ro) * 16 + ao) << 15) + (od << 10) + (oh << 5) + ow] =
          pre[j] * scale;
    }
  }
}

extern "C" void kernel_launch(void* const* d_in, const int* in_sizes, int n_in,
                              void* d_out, int out_size, void* d_ws, size_t ws_size,
                              hipStream_t stream) {
  (void)in_sizes; (void)n_in; (void)out_size; (void)d_ws; (void)ws_size;
  const float* x = (const float*)d_in[0];
  const float* w = (const float*)d_in[1];
  const float* b = (const float*)d_in[2];
  float* out = (float*)d_out;

  const size_t shmem = (size_t)(XS_N + WS_N + VS_N) * sizeof(float);
  // grid: b(4) x od(32) x oh(32); block: 32 waves = (par x i x o) GEMM tiles
  deconv_caps_routing<<<dim3(4 * 32 * 32), dim3(1024), shmem, stream>>>(x, w, b, out);
}